// CWALayerv3_79190607004225
// MI455X (gfx1250) — compile-verified
//
#include <hip/hip_runtime.h>

typedef float v2f __attribute__((ext_vector_type(2)));
typedef float v8f __attribute__((ext_vector_type(8)));

#define NB 4
#define NC 256
#define HP 64
#define PLANE (HP * HP)  // 4096
#define KSPLIT 2
#define KCHUNK (PLANE / KSPLIT)  // 2048

// ---------------------------------------------------------------------------
// Kernel 1: AvgPool2d(8) on [4,256,512,512] -> xp [4,256,64,64]
// One block per output row (b,c,oh). 512 threads = 512 input columns.
// Perfectly coalesced streaming of the 1.07 GB input (HBM-bound, ~46 us at
// 23.3 TB/s — this pass is the runtime floor for the whole problem).
// ---------------------------------------------------------------------------
__global__ __launch_bounds__(512) void avgpool8_kernel(
    const float* __restrict__ x, float* __restrict__ xp) {
  __shared__ float lds[512];
  const int tid = threadIdx.x;
  const int bid = blockIdx.x;  // ((b*NC + c)*64 + oh)

  const float* px = x + (long)bid * 4096 + tid;  // 8 rows x 512 cols region
  float s = 0.f;
#pragma unroll
  for (int r = 0; r < 8; ++r) s += px[r * 512];
  lds[tid] = s;
  __syncthreads();
  if (tid < 64) {
    float t = 0.f;
#pragma unroll
    for (int j = 0; j < 8; ++j) t += lds[tid * 8 + j];
    xp[(long)bid * 64 + tid] = t * (1.0f / 64.0f);
  }
}

// ---------------------------------------------------------------------------
// Kernel 2: fused relu(dwconv1) -> relu(dwconv2 + xp) -> InstanceNorm, per
// (b,c) 64x64 plane, entirely in LDS (zero-padded 66x66 tiles).
// Writes normalized features f [4,256,4096]. All traffic is L2-resident.
// ---------------------------------------------------------------------------
__global__ __launch_bounds__(256) void fused_resblock_inorm_kernel(
    const float* __restrict__ xp, const float* __restrict__ w1,
    const float* __restrict__ w2, const float* __restrict__ gamma,
    const float* __restrict__ beta, float* __restrict__ f) {
  const int tid = threadIdx.x;
  const int pc = blockIdx.x;  // b*NC + c
  const int c = pc & (NC - 1);

  __shared__ float XP[66 * 66];
  __shared__ float T1[66 * 66];
  __shared__ float red[256];

  for (int i = tid; i < 66 * 66; i += 256) { XP[i] = 0.f; T1[i] = 0.f; }
  __syncthreads();

  const float* plane = xp + (long)pc * PLANE;
#pragma unroll
  for (int i = 0; i < 16; ++i) {
    int p = i * 256 + tid;
    XP[((p >> 6) + 1) * 66 + (p & 63) + 1] = plane[p];
  }

  float a1[9], a2[9];
#pragma unroll
  for (int j = 0; j < 9; ++j) { a1[j] = w1[c * 9 + j]; a2[j] = w2[c * 9 + j]; }
  __syncthreads();

  // conv1 + relu -> T1 (padded interior)
#pragma unroll
  for (int i = 0; i < 16; ++i) {
    int p = i * 256 + tid;
    int r = (p >> 6) + 1, cc = (p & 63) + 1;
    const float* q = &XP[(r - 1) * 66 + cc - 1];
    float v = q[0] * a1[0] + q[1] * a1[1] + q[2] * a1[2] +
              q[66] * a1[3] + q[67] * a1[4] + q[68] * a1[5] +
              q[132] * a1[6] + q[133] * a1[7] + q[134] * a1[8];
    T1[r * 66 + cc] = fmaxf(v, 0.f);
  }
  __syncthreads();

  // conv2 + residual + relu, kept in registers; accumulate stats
  float o[16];
  float s = 0.f, s2 = 0.f;
#pragma unroll
  for (int i = 0; i < 16; ++i) {
    int p = i * 256 + tid;
    int r = (p >> 6) + 1, cc = (p & 63) + 1;
    const float* q = &T1[(r - 1) * 66 + cc - 1];
    float v = q[0] * a2[0] + q[1] * a2[1] + q[2] * a2[2] +
              q[66] * a2[3] + q[67] * a2[4] + q[68] * a2[5] +
              q[132] * a2[6] + q[133] * a2[7] + q[134] * a2[8] +
              XP[r * 66 + cc];
    v = fmaxf(v, 0.f);
    o[i] = v;
    s += v;
    s2 += v * v;
  }

  red[tid] = s;
  __syncthreads();
  for (int off = 128; off > 0; off >>= 1) {
    if (tid < off) red[tid] += red[tid + off];
    __syncthreads();
  }
  const float sum1 = red[0];
  __syncthreads();
  red[tid] = s2;
  __syncthreads();
  for (int off = 128; off > 0; off >>= 1) {
    if (tid < off) red[tid] += red[tid + off];
    __syncthreads();
  }
  const float sum2 = red[0];

  const float mean = sum1 * (1.0f / PLANE);
  const float var = sum2 * (1.0f / PLANE) - mean * mean;  // biased
  const float sc = rsqrtf(var + 1e-5f) * gamma[c];
  const float bt = beta[c];

  float* fp = f + (long)pc * PLANE;
#pragma unroll
  for (int i = 0; i < 16; ++i) {
    int p = i * 256 + tid;
    fp[p] = (o[i] - mean) * sc + bt;
  }
}

// ---------------------------------------------------------------------------
// Kernel 3: batched Gram S = F F^T via V_WMMA_F32_16X16X4_F32 (fp32 matrix
// path, matching the fp32 reference). One wave per (16x16 tile, K-chunk):
//   4 batches x 256 tiles x 2 K-chunks = 2048 waves (256 blocks x 8 waves).
// 4 independent accumulator chains (dep distance 4 WMMAs) hide XDL latency
// even at ~1-2 waves/SIMD. Partial tiles land in 2 separate buffers so the
// combine is deterministic (no float atomics).
// Per ISA layouts:
//   A 16x4 : lane<16 -> M=lane, {v0,v1}={K0,K1}; lane>=16 -> {K2,K3}
//   B 4x16 : lane<16 -> N=lane, {v0,v1}={K0,K1} rows; lane>=16 -> {K2,K3}
//   C/D    : element vi -> row vi + 8*(lane>>4), col lane&15
// ---------------------------------------------------------------------------
__global__ __launch_bounds__(256) void gram_wmma_kernel(
    const float* __restrict__ f, float* __restrict__ sgpart) {
  const int tid = threadIdx.x;
  const int lane = tid & 31;
  const int wv = tid >> 5;
  const int gw = blockIdx.x * 8 + wv;  // 0..2047
  const int kc = gw >> 10;             // K-chunk 0/1
  const int rest = gw & 1023;
  const int b = rest >> 8;
  const int tile = rest & 255;
  const int m0 = (tile >> 4) << 4;
  const int n0 = (tile & 15) << 4;
  const int half = lane >> 4;
  const int lr = lane & 15;
  const int k0 = kc * KCHUNK;

  const float* F = f + (long)b * NC * PLANE;
  const float* pa = F + (long)(m0 + lr) * PLANE + k0 + 2 * half;
  const float* pb = F + (long)(n0 + lr) * PLANE + k0 + 2 * half;

  v8f c0 = {}, c1 = {}, c2 = {}, c3 = {};
#pragma unroll 2
  for (int k = 0; k < KCHUNK; k += 16) {
    v2f a0 = *(const v2f*)(pa + k);
    v2f b0 = *(const v2f*)(pb + k);
    c0 = __builtin_amdgcn_wmma_f32_16x16x4_f32(false, a0, false, b0,
                                               (short)0, c0, false, false);
    v2f a1v = *(const v2f*)(pa + k + 4);
    v2f b1v = *(const v2f*)(pb + k + 4);
    c1 = __builtin_amdgcn_wmma_f32_16x16x4_f32(false, a1v, false, b1v,
                                               (short)0, c1, false, false);
    v2f a2v = *(const v2f*)(pa + k + 8);
    v2f b2v = *(const v2f*)(pb + k + 8);
    c2 = __builtin_amdgcn_wmma_f32_16x16x4_f32(false, a2v, false, b2v,
                                               (short)0, c2, false, false);
    v2f a3v = *(const v2f*)(pa + k + 12);
    v2f b3v = *(const v2f*)(pb + k + 12);
    c3 = __builtin_amdgcn_wmma_f32_16x16x4_f32(false, a3v, false, b3v,
                                               (short)0, c3, false, false);
  }

  float* out = sgpart + ((long)kc * NB + b) * NC * NC;
#pragma unroll
  for (int vi = 0; vi < 8; ++vi) {
    int row = m0 + vi + 8 * half;
    int col = n0 + lr;
    out[row * NC + col] = (c0[vi] + c1[vi]) + (c2[vi] + c3[vi]);
  }
}

// ---------------------------------------------------------------------------
// Kernel 4: combine the two K-partials (deterministic), sigmoid, diagonal
// mask, row mean -> out [4,256]. 2 MB of L2-resident traffic, trivial.
// ---------------------------------------------------------------------------
__global__ __launch_bounds__(256) void finalize_kernel(
    const float* __restrict__ sgpart, float* __restrict__ out) {
  __shared__ float red[256];
  const int tid = threadIdx.x;
  const int row = blockIdx.x;  // b*NC + c
  const int b = row >> 8;
  const int c = row & (NC - 1);

  const long idx = (long)b * NC * NC + (long)c * NC + tid;
  float v = sgpart[idx] + sgpart[(long)NB * NC * NC + idx];
  float sgm = 1.0f / (1.0f + __expf(-v));
  if (tid == c) sgm = 0.f;  // 1 - eye mask
  red[tid] = sgm;
  __syncthreads();
  for (int off = 128; off > 0; off >>= 1) {
    if (tid < off) red[tid] += red[tid + off];
    __syncthreads();
  }
  if (tid == 0) out[row] = red[0] * (1.0f / NC);
}

extern "C" void kernel_launch(void* const* d_in, const int* in_sizes, int n_in,
                              void* d_out, int out_size, void* d_ws,
                              size_t ws_size, hipStream_t stream) {
  const float* x = (const float*)d_in[0];
  const float* w1 = (const float*)d_in[1];
  const float* w2 = (const float*)d_in[2];
  const float* gamma = (const float*)d_in[3];
  const float* beta = (const float*)d_in[4];
  float* out = (float*)d_out;

  float* xp = (float*)d_ws;                     // 16 MB
  float* f = xp + (size_t)NB * NC * PLANE;      // 16 MB
  float* sgpart = f + (size_t)NB * NC * PLANE;  // 2 x 1 MB (K-split partials)

  avgpool8_kernel<<<NB * NC * 64, 512, 0, stream>>>(x, xp);
  fused_resblock_inorm_kernel<<<NB * NC, 256, 0, stream>>>(xp, w1, w2, gamma,
                                                           beta, f);
  gram_wmma_kernel<<<256, 256, 0, stream>>>(f, sgpart);
  finalize_kernel<<<NB * NC, 256, 0, stream>>>(sgpart, out);
}